// YOLOv1Loss_88776974008933
// MI455X (gfx1250) — compile-verified
//
#include <hip/hip_runtime.h>
#include <hip/hip_bf16.h>
#include <math.h>

#define NUM_CLASSES 80
#define S_GRID      14
#define CH          90            // 10 + NUM_CLASSES
#define LAMBDA_COORD 5.0f
#define LAMBDA_NOOBJ 0.5f
#define EPS_IOU 1e-6f
#define EPS_WH  1e-8f

typedef float v2f __attribute__((ext_vector_type(2)));
typedef float v8f __attribute__((ext_vector_type(8)));

// Exact f32 wave32 sum using V_WMMA_F32_16X16X4_F32.
// A[m][0] = acc(lane m), A[m][2] = acc(lane m+16), A[*][1]=A[*][3]=0, B = ones.
// => D[m][n] = acc[m] + acc[m+16].  D reg r: lanes 0-15 hold row r, lanes 16-31 row r+8.
// Per-lane sum of the 8 D regs gives half-sums; shfl_xor(16) completes the total.
__device__ __forceinline__ float wave_sum_wmma(float v) {
    v2f a; a[0] = v;    a[1] = 0.0f;
    v2f b; b[0] = 1.0f; b[1] = 1.0f;
    v8f c = {};
    v8f d = __builtin_amdgcn_wmma_f32_16x16x4_f32(
        /*neg_a=*/false, a, /*neg_b=*/false, b,
        /*c_mod=*/(short)0, c, /*reuse_a=*/false, /*reuse_b=*/false);
    float s = ((d[0] + d[1]) + (d[2] + d[3])) + ((d[4] + d[5]) + (d[6] + d[7]));
    s += __shfl_xor(s, 16, 32);
    return s;
}

__global__ void yolo_init(float* __restrict__ ws) {
    if (threadIdx.x < 16) ws[threadIdx.x] = 0.0f;
}

__global__ __launch_bounds__(256) void yolo_partial(
        const float* __restrict__ pred, const float* __restrict__ target,
        float* __restrict__ ws, int ncells) {
    const int cell = blockIdx.x * blockDim.x + threadIdx.x;

    float acc_coord = 0.f, acc_conf = 0.f, acc_no1 = 0.f, acc_nno = 0.f,
          acc_no2 = 0.f, acc_cls1 = 0.f, acc_cls2 = 0.f, acc_n1 = 0.f,
          acc_n2 = 0.f, acc_bad = 0.f;

    if (cell < ncells) {
        const float2* p2 = reinterpret_cast<const float2*>(pred)   + (size_t)cell * (CH / 2);
        const float2* t2 = reinterpret_cast<const float2*>(target) + (size_t)cell * (CH / 2);
        __builtin_prefetch(p2, 0, 0);   // global_prefetch_b8
        __builtin_prefetch(t2, 0, 0);

        // Box channels: pred ch0-9, target ch0-9 (ch5-9 only feed the NaN check).
        float2 pA = p2[0], pB = p2[1], pC = p2[2], pD = p2[3], pE = p2[4];
        float2 tA = t2[0], tB = t2[1], tC = t2[2], tD = t2[3], tE = t2[4];

        int bad = 0;
        float clsdiff = 0.f;
        #pragma unroll
        for (int i = 5; i < CH / 2; ++i) {          // class channels 10..89
            float2 pv = p2[i], tv = t2[i];
            float d0 = pv.x - tv.x, d1 = pv.y - tv.y;
            clsdiff += d0 * d0 + d1 * d1;
            bad |= (pv.x != pv.x) | (pv.y != pv.y) | (tv.x != tv.x) | (tv.y != tv.y);
        }
        bad |= (pA.x != pA.x) | (pA.y != pA.y) | (pB.x != pB.x) | (pB.y != pB.y);
        bad |= (pC.x != pC.x) | (pC.y != pC.y) | (pD.x != pD.x) | (pD.y != pD.y);
        bad |= (pE.x != pE.x) | (pE.y != pE.y);
        bad |= (tA.x != tA.x) | (tA.y != tA.y) | (tB.x != tB.x) | (tB.y != tB.y);
        bad |= (tC.x != tC.x) | (tC.y != tC.y) | (tD.x != tD.x) | (tD.y != tD.y);
        bad |= (tE.x != tE.x) | (tE.y != tE.y);

        const float px1 = pA.x, py1 = pA.y, pw1 = pB.x, ph1 = pB.y, pc1 = pC.x;
        const float px2 = pC.y, py2 = pD.x, pw2 = pD.y, ph2 = pE.x, pc2 = pE.y;
        const float tx  = tA.x, ty  = tA.y, tw  = tB.x, th  = tB.y, tcf = tC.x;

        const int   cy   = (cell / S_GRID) % S_GRID;
        const int   cx   = cell % S_GRID;
        const float invS = 1.0f / (float)S_GRID;
        const float tix  = ((float)cx + tx) * invS;
        const float tiy  = ((float)cy + ty) * invS;

        auto iou_vs_t = [&](float bx, float by, float bw, float bh) -> float {
            float ix = ((float)cx + bx) * invS;
            float iy = ((float)cy + by) * invS;
            float x0 = fmaxf(ix - bw * 0.5f, tix - tw * 0.5f);
            float y0 = fmaxf(iy - bh * 0.5f, tiy - th * 0.5f);
            float x1 = fminf(ix + bw * 0.5f, tix + tw * 0.5f);
            float y1 = fminf(iy + bh * 0.5f, tiy + th * 0.5f);
            float inter = fmaxf(x1 - x0, 0.f) * fmaxf(y1 - y0, 0.f);
            float uni   = bw * bh + tw * th - inter;
            return inter / (uni + EPS_IOU);
        };
        const float iou1 = iou_vs_t(px1, py1, pw1, ph1);
        const float iou2 = iou_vs_t(px2, py2, pw2, ph2);

        const bool  obj   = (tcf == 1.0f);
        const bool  noobj = (tcf == 0.0f);
        const float m1 = (obj && (iou1 >= iou2)) ? 1.f : 0.f;
        const float m2 = (obj && (iou1 <  iou2)) ? 1.f : 0.f;

        auto coord_part = [&](float bx, float by, float bw, float bh) -> float {
            float dx = bx - tx, dy = by - ty;
            float sw = sqrtf(fmaxf(bw, EPS_WH)) - sqrtf(fmaxf(tw, EPS_WH));
            float sh = sqrtf(fmaxf(bh, EPS_WH)) - sqrtf(fmaxf(th, EPS_WH));
            return dx * dx + dy * dy + sw * sw + sh * sh;
        };

        acc_coord = m1 * coord_part(px1, py1, pw1, ph1)
                  + m2 * coord_part(px2, py2, pw2, ph2);
        acc_conf  = m1 * ((pc1 - iou1) * (pc1 - iou1) + pc2 * pc2)
                  + m2 * ((pc2 - iou2) * (pc2 - iou2) + pc1 * pc1);
        acc_no1   = noobj ? pc1 * pc1 : 0.f;
        acc_nno   = noobj ? 1.f : 0.f;
        acc_no2   = pc2 * pc2;                 // mean over all cells, done in finalize
        acc_cls1  = m1 * clsdiff;
        acc_cls2  = m2 * clsdiff;
        acc_n1    = m1;
        acc_n2    = m2;
        acc_bad   = bad ? 1.f : 0.f;
    }

    // EXEC reconverged to all-ones here: WMMA-based wave32 reductions.
    acc_coord = wave_sum_wmma(acc_coord);
    acc_conf  = wave_sum_wmma(acc_conf);
    acc_no1   = wave_sum_wmma(acc_no1);
    acc_nno   = wave_sum_wmma(acc_nno);
    acc_no2   = wave_sum_wmma(acc_no2);
    acc_cls1  = wave_sum_wmma(acc_cls1);
    acc_cls2  = wave_sum_wmma(acc_cls2);
    acc_n1    = wave_sum_wmma(acc_n1);
    acc_n2    = wave_sum_wmma(acc_n2);
    acc_bad   = wave_sum_wmma(acc_bad);

    if ((threadIdx.x & 31) == 0) {
        atomicAdd(&ws[0], acc_coord);
        atomicAdd(&ws[1], acc_conf);
        atomicAdd(&ws[2], acc_no1);
        atomicAdd(&ws[3], acc_nno);
        atomicAdd(&ws[4], acc_no2);
        atomicAdd(&ws[5], acc_cls1);
        atomicAdd(&ws[6], acc_cls2);
        atomicAdd(&ws[7], acc_n1);
        atomicAdd(&ws[8], acc_n2);
        if (acc_bad > 0.f) atomicAdd(&ws[9], 1.f);
    }
}

__global__ void yolo_finalize(const float* __restrict__ ws, float* __restrict__ out,
                              int ncells, int batch) {
    float coord = LAMBDA_COORD * ws[0];
    float conf  = ws[1];
    float nno   = ws[3];
    float noobj1 = ws[2] / fmaxf(nno, 1.0f);
    float noobj2 = ws[4] / (float)ncells;
    conf += (nno > 0.f) ? LAMBDA_NOOBJ * (noobj1 + noobj2) : 0.f;
    float n1 = ws[7], n2 = ws[8];
    float cls = ((n1 > 0.f) ? ws[5] / fmaxf(n1 * (float)NUM_CLASSES, 1.f) : 0.f)
              + ((n2 > 0.f) ? ws[6] / fmaxf(n2 * (float)NUM_CLASSES, 1.f) : 0.f);
    float total = (coord + conf + cls) / (float)batch;
    out[0] = (ws[9] > 0.f) ? 0.f : total;
}

extern "C" void kernel_launch(void* const* d_in, const int* in_sizes, int n_in,
                              void* d_out, int out_size, void* d_ws, size_t ws_size,
                              hipStream_t stream) {
    const float* pred   = (const float*)d_in[0];
    const float* target = (const float*)d_in[1];
    float*       out    = (float*)d_out;
    float*       ws     = (float*)d_ws;

    const int ncells = in_sizes[0] / CH;                 // 2048*14*14 = 401408
    const int batch  = ncells / (S_GRID * S_GRID);       // 2048

    yolo_init<<<1, 32, 0, stream>>>(ws);

    const int threads = 256;
    const int blocks  = (ncells + threads - 1) / threads;  // 1568, exact
    yolo_partial<<<blocks, threads, 0, stream>>>(pred, target, ws, ncells);

    yolo_finalize<<<1, 1, 0, stream>>>(ws, out, ncells, batch);
}